// HeteroGCN_3246995275924
// MI455X (gfx1250) — compile-verified
//
#include <hip/hip_runtime.h>
#include <hip/hip_bf16.h>

#define N_NODES 100000
#define N_EDGES 3200000
#define D_FEAT  256
#define LEAKY   0.2f

typedef float v2f __attribute__((ext_vector_type(2)));
typedef float v8f __attribute__((ext_vector_type(8)));

// ---------------- Stage 1: degree histogram ----------------
__global__ void k_zero(int* __restrict__ deg, int* __restrict__ cursor, int n) {
    int i = blockIdx.x * blockDim.x + threadIdx.x;
    if (i < n) { deg[i] = 0; cursor[i] = 0; }
}

__global__ void k_degree(const int* __restrict__ row, int* __restrict__ deg, int E) {
    int e = blockIdx.x * blockDim.x + threadIdx.x;
    if (e < E) atomicAdd(&deg[row[e]], 1);
}

// ---------------- Stage 2: exclusive scan + dinv (single block) ----------------
__global__ __launch_bounds__(1024) void k_scan(const int* __restrict__ deg,
                                               int* __restrict__ offsets,
                                               float* __restrict__ dinv, int n) {
    __shared__ int part[1024];
    const int t  = threadIdx.x;
    const int CH = (n + 1023) / 1024;      // 98 for n=100000
    const int start = t * CH;
    int s = 0;
    for (int i = 0; i < CH; ++i) {
        int idx = start + i;
        if (idx < n) s += deg[idx];
    }
    part[t] = s;
    __syncthreads();
    // Hillis-Steele inclusive scan over 1024 partials
    for (int off = 1; off < 1024; off <<= 1) {
        int v = (t >= off) ? part[t - off] : 0;
        __syncthreads();
        part[t] += v;
        __syncthreads();
    }
    int run = (t == 0) ? 0 : part[t - 1];  // exclusive prefix for this chunk
    for (int i = 0; i < CH; ++i) {
        int idx = start + i;
        if (idx < n) {
            offsets[idx] = run;
            int d = deg[idx];
            run += d;
            dinv[idx] = (d > 0) ? rsqrtf((float)d) : 0.0f;
        }
    }
}

// ---------------- Stage 3: CSR scatter ----------------
__global__ void k_scatter(const int* __restrict__ row, const int* __restrict__ col,
                          const int* __restrict__ offsets, int* __restrict__ cursor,
                          int* __restrict__ edge_dst, int E) {
    int e = blockIdx.x * blockDim.x + threadIdx.x;
    if (e < E) {
        int r = row[e];
        int p = atomicAdd(&cursor[r], 1);
        edge_dst[offsets[r] + p] = col[e];
    }
}

// ---------------- Stage 4: gather SpMM (no float atomics, coalesced x reads) ----
__global__ __launch_bounds__(256) void k_spmm(const float* __restrict__ x,
                                              const int* __restrict__ edge_dst,
                                              const int* __restrict__ offsets,
                                              const int* __restrict__ deg,
                                              const float* __restrict__ dinv,
                                              float* __restrict__ h) {
    __shared__ int   s_c[256];
    __shared__ float s_w[256];
    const int node = blockIdx.x;
    const int f    = threadIdx.x;            // feature column 0..255
    const int off  = offsets[node];
    const int cnt  = deg[node];
    float acc = 0.0f;
    for (int base = 0; base < cnt; base += 256) {
        int rem = cnt - base;
        int msz = rem < 256 ? rem : 256;
        if (f < msz) {
            int c  = edge_dst[off + base + f];
            s_c[f] = c;
            s_w[f] = dinv[c];
        }
        __syncthreads();
        for (int e = 0; e < msz; ++e) {
            acc = fmaf(s_w[e], x[(size_t)s_c[e] * D_FEAT + f], acc);
        }
        __syncthreads();
    }
    h[(size_t)node * D_FEAT + f] = dinv[node] * acc;
}

// ---------------- Stage 5: in-place GEMM (h @ W + b, leaky-ReLU) via fp32 WMMA ----
// Block = 256 threads (8 waves), handles 16 rows x 256 cols in place.
// W streamed through LDS in K-panels of 32, stored transposed (stride 36 ->
// conflict-free b64 B-fragment loads). H slab stride 260 -> conflict-free A loads.
#define KP        32
#define WT_STRIDE 36
#define H_STRIDE  260

__global__ __launch_bounds__(256) void k_gemm(float* __restrict__ h,
                                              const float* __restrict__ Wm,
                                              const float* __restrict__ bias) {
    __shared__ float sWt[256 * WT_STRIDE];   // [ncol][kk]   36 KB
    __shared__ float sH [16  * H_STRIDE];    // [m][k]       16.6 KB
    const int  tid  = threadIdx.x;
    const int  wave = tid >> 5;
    const int  lane = tid & 31;
    const long rowBase = (long)blockIdx.x * 16;

    // Stage own 16 rows of h into LDS (before any in-place write)
    for (int i = tid; i < 16 * 64; i += 256) {
        int r = i >> 6, c4 = i & 63;
        float4 v = *(const float4*)(h + (rowBase + r) * 256 + c4 * 4);
        *(float4*)(sH + r * H_STRIDE + c4 * 4) = v;
    }

    v8f acc0 = {};
    v8f acc1 = {};
    const int m   = lane & 15;               // A-fragment row
    const int n   = lane & 15;               // B/C-fragment column
    const int kb  = (lane >> 4) << 1;        // K sub-offset: 0 (lanes 0-15) / 2 (16-31)
    const int ct0 = wave * 2;                // this wave's two 16-col tiles
    const int ct1 = wave * 2 + 1;

    for (int kp = 0; kp < 256; kp += KP) {
        __syncthreads();                      // previous panel fully consumed (also covers sH stage)
        // Stage W panel transposed: sWt[n][kk] = W[kp+kk][n]
        for (int i = tid; i < KP * 256; i += 256) {
            int kk = i >> 8, c = i & 255;
            sWt[c * WT_STRIDE + kk] = Wm[(kp + kk) * 256 + c];
        }
        __syncthreads();
#pragma unroll
        for (int k0 = 0; k0 < KP; k0 += 4) {
            v2f a  = *(const v2f*)(sH  + m * H_STRIDE + kp + k0 + kb);
            v2f b0 = *(const v2f*)(sWt + (ct0 * 16 + n) * WT_STRIDE + k0 + kb);
            v2f b1 = *(const v2f*)(sWt + (ct1 * 16 + n) * WT_STRIDE + k0 + kb);
            acc0 = __builtin_amdgcn_wmma_f32_16x16x4_f32(false, a, false, b0,
                                                         (short)0, acc0, false, false);
            acc1 = __builtin_amdgcn_wmma_f32_16x16x4_f32(false, a, false, b1,
                                                         (short)0, acc1, false, false);
        }
    }

    // Write out: C/D VGPR r -> row (r + 8*(lane>=16)), col = n
    const int mrow = (lane >> 4) << 3;
    const int nc0 = ct0 * 16 + n;
    const int nc1 = ct1 * 16 + n;
    const float b0v = bias[nc0];
    const float b1v = bias[nc1];
#pragma unroll
    for (int r = 0; r < 8; ++r) {
        float v0 = acc0[r] + b0v; v0 = (v0 >= 0.0f) ? v0 : LEAKY * v0;
        float v1 = acc1[r] + b1v; v1 = (v1 >= 0.0f) ? v1 : LEAKY * v1;
        h[(rowBase + mrow + r) * 256 + nc0] = v0;
        h[(rowBase + mrow + r) * 256 + nc1] = v1;
    }
}

// ---------------- Launch ----------------
extern "C" void kernel_launch(void* const* d_in, const int* in_sizes, int n_in,
                              void* d_out, int out_size, void* d_ws, size_t ws_size,
                              hipStream_t stream) {
    const float* x    = (const float*)d_in[0];
    const int*   ei   = (const int*)d_in[1];
    const float* Wm   = (const float*)d_in[2];
    const float* bias = (const float*)d_in[3];
    float*       out  = (float*)d_out;

    const int* row = ei;             // edge_index[0]
    const int* col = ei + N_EDGES;   // edge_index[1]

    // Workspace partition (~14.4 MB)
    int*   deg      = (int*)d_ws;
    int*   offsets  = deg + N_NODES;
    int*   cursor   = offsets + N_NODES;
    float* dinv     = (float*)(cursor + N_NODES);
    int*   edge_dst = (int*)(dinv + N_NODES);

    k_zero   <<<(N_NODES + 255) / 256, 256, 0, stream>>>(deg, cursor, N_NODES);
    k_degree <<<(N_EDGES + 255) / 256, 256, 0, stream>>>(row, deg, N_EDGES);
    k_scan   <<<1, 1024, 0, stream>>>(deg, offsets, dinv, N_NODES);
    k_scatter<<<(N_EDGES + 255) / 256, 256, 0, stream>>>(row, col, offsets, cursor,
                                                         edge_dst, N_EDGES);
    k_spmm   <<<N_NODES, 256, 0, stream>>>(x, edge_dst, offsets, deg, dinv, out);
    k_gemm   <<<N_NODES / 16, 256, 0, stream>>>(out, Wm, bias);
}